// GraphAttentionLayer_49460843380764
// MI455X (gfx1250) — compile-verified
//
#include <hip/hip_runtime.h>
#include <hip/hip_bf16.h>

// Problem constants (from reference setup_inputs)
#define BB   2
#define TT   96
#define NN   64
#define CIN  16
#define COUT 32
#define PP   24      // PATCH_SIZE
#define NP   4       // T / P
#define ALPHA 0.2f

#define M_TOTAL (BB * TT * NN)              // 12288 rows of the flat h-GEMM
#define H_ELEMS (M_TOTAL * COUT)            // 393216 floats
#define ATT_ELEMS (BB * NN * NN * TT * COUT) // 25165824 floats
#define HP_ELEMS (BB * NN * TT * COUT)      // 393216 floats

typedef __attribute__((ext_vector_type(2))) float v2f;
typedef __attribute__((ext_vector_type(8))) float v8f;

// ---------------------------------------------------------------------------
// Kernel 1: h = x_flat(12288x16) @ W(16x32) via V_WMMA_F32_16X16X4_F32.
// One wave -> one 16x32 tile of h (4 K-steps of 4, two 16-col halves).
// f32 WMMA VGPR layouts (ISA 7.12.2):
//   A 16x4 : lane 0-15 M=lane, VGPR0 K=(0|2 by lane half), VGPR1 K=(1|3)
//   B 4x16 : lane 0-15 N=lane, same K striping (B is A's transpose shape)
//   C 16x16: VGPR r holds M=r (lanes 0-15) / M=r+8 (lanes 16-31), N=lane&15
// ---------------------------------------------------------------------------
__global__ void gat_h_wmma_kernel(const float* __restrict__ x,
                                  const float* __restrict__ W,
                                  float* __restrict__ h) {
  const int gwave = (blockIdx.x * blockDim.x + threadIdx.x) >> 5;
  const int lane  = threadIdx.x & 31;
  const int m0    = gwave * 16;
  if (m0 >= M_TOTAL) return;

  const int half = lane >> 4;   // 0: lanes 0-15, 1: lanes 16-31
  const int l    = lane & 15;

  v8f c0 = {};   // cols 0..15
  v8f c1 = {};   // cols 16..31

#pragma unroll
  for (int kk = 0; kk < 4; ++kk) {
    const int kb = kk * 4 + 2 * half;       // K base for this lane half
    v2f a, b0, b1;
    a.x  = x[(m0 + l) * CIN + kb];
    a.y  = x[(m0 + l) * CIN + kb + 1];
    b0.x = W[kb * COUT + l];
    b0.y = W[(kb + 1) * COUT + l];
    b1.x = W[kb * COUT + 16 + l];
    b1.y = W[(kb + 1) * COUT + 16 + l];
    c0 = __builtin_amdgcn_wmma_f32_16x16x4_f32(false, a, false, b0,
                                               (short)0, c0, false, false);
    c1 = __builtin_amdgcn_wmma_f32_16x16x4_f32(false, a, false, b1,
                                               (short)0, c1, false, false);
  }

#pragma unroll
  for (int r = 0; r < 8; ++r) {
    const int m = r + 8 * half;
    h[(m0 + m) * COUT + l]      = c0[r];
    h[(m0 + m) * COUT + 16 + l] = c1[r];
  }
}

// ---------------------------------------------------------------------------
// Kernel 2: attention + row-sum S.
// Block = 256 threads (8 waves). Each block owns (b,s,p, i-group of 8).
// LDS holds h[b,s,0..63,p,0..31] (8 KB). Wave w -> i = igroup*8 + w,
// lane = channel c. Loop j: e = leaky(h_i*h_j)*adj[i,j], softmax over the
// 32 lanes (= channel dim), store attention[b,i,j,t=p*4+s,c], accumulate S.
// ---------------------------------------------------------------------------
__global__ void gat_attn_kernel(const float* __restrict__ h,
                                const float* __restrict__ adj,
                                float* __restrict__ att,
                                float* __restrict__ S) {
  __shared__ float hs[NN * COUT];           // 8 KB

  const int g      = blockIdx.x;            // 192 * 8 blocks
  const int igroup = g & 7;
  const int bsp    = g >> 3;                // 0..191
  const int p      = bsp % PP;
  const int s      = (bsp / PP) % NP;
  const int b      = bsp / (PP * NP);

  const int tid  = threadIdx.x;
  const int lane = tid & 31;                // channel c
  const int wave = tid >> 5;                // 0..7
  const int i    = igroup * 8 + wave;

  // Stage all 64 node rows for this (b,s,p): h[(((b*NP+s)*NN+j)*PP+p)*COUT+c]
  const long hbase = ((long)(b * NP + s) * NN) * PP * COUT + (long)p * COUT;
  for (int idx = tid; idx < NN * COUT; idx += 256) {
    const int j = idx >> 5;
    const int c = idx & 31;
    hs[idx] = h[hbase + (long)j * (PP * COUT) + c];
  }
  __syncthreads();

  const float hi = hs[i * COUT + lane];
  const int   t  = p * NP + s;              // attention's time index
  float* attrow  = att + (((long)(b * NN + i) * NN) * TT + t) * COUT + lane;

  float sumj = 0.0f;
#pragma unroll 4
  for (int j = 0; j < NN; ++j) {
    float v = hi * hs[j * COUT + lane];
    v = (v >= 0.0f) ? v : ALPHA * v;        // LeakyReLU
    v *= adj[i * NN + j];

    // softmax over the 32 channels (one per lane)
    float mx = v;
#pragma unroll
    for (int off = 16; off >= 1; off >>= 1)
      mx = fmaxf(mx, __shfl_xor(mx, off, 32));
    const float ex = __expf(v - mx);
    float sm = ex;
#pragma unroll
    for (int off = 16; off >= 1; off >>= 1)
      sm += __shfl_xor(sm, off, 32);

    const float a = ex / sm;
    attrow[(long)j * (TT * COUT)] = a;      // coalesced 128B per wave
    sumj += a;
  }

  // S is (B, N, T, C); each wave owns a unique row -> plain store
  S[(((long)(b * NN + i) * TT) + t) * COUT + lane] = sumj;
}

// ---------------------------------------------------------------------------
// Kernel 3: h_prime = elu(S * h_flat), h_flat t-mapping is t = s*P + p.
// ---------------------------------------------------------------------------
__global__ void gat_hprime_kernel(const float* __restrict__ h,
                                  const float* __restrict__ S,
                                  float* __restrict__ hp) {
  const int idx = blockIdx.x * blockDim.x + threadIdx.x;
  if (idx >= HP_ELEMS) return;
  const int c  = idx & 31;
  const int t  = (idx >> 5) % TT;
  const int n  = (idx >> 5) / TT % NN;
  const int b  = idx / (COUT * TT * NN);
  const int s  = t / PP;
  const int pp = t % PP;
  const float hv  = h[((((long)(b * NP + s) * NN + n) * PP) + pp) * COUT + c];
  const float val = S[idx] * hv;
  hp[idx] = (val > 0.0f) ? val : (__expf(val) - 1.0f);   // ELU (alpha=1)
}

// ---------------------------------------------------------------------------
extern "C" void kernel_launch(void* const* d_in, const int* in_sizes, int n_in,
                              void* d_out, int out_size, void* d_ws, size_t ws_size,
                              hipStream_t stream) {
  const float* x   = (const float*)d_in[0];   // (B,T,N,Cin)
  const float* adj = (const float*)d_in[1];   // (N,N)
  const float* W   = (const float*)d_in[2];   // (Cin,Cout)
  // d_in[3] == W_c: unused by the reference forward

  float* out = (float*)d_out;
  float* att = out;                 // first output: (B,N,N,T,C)
  float* hp  = out + ATT_ELEMS;     // second output: (B,N,T,C)

  float* h = (float*)d_ws;          // (B,NP,N,P,Cout)  = 1.5 MB
  float* S = h + H_ELEMS;           // (B,N,T,Cout)     = 1.5 MB

  // Kernel 1: 768 waves (16 rows each), 8 waves/block -> 96 blocks
  gat_h_wmma_kernel<<<96, 256, 0, stream>>>(x, W, h);

  // Kernel 2: 192 (b,s,p) groups x 8 i-groups = 1536 blocks of 256
  gat_attn_kernel<<<BB * NP * PP * 8, 256, 0, stream>>>(h, adj, att, S);

  // Kernel 3: 393216 elems / 256 = 1536 blocks
  gat_hprime_kernel<<<(HP_ELEMS + 255) / 256, 256, 0, stream>>>(h, S, hp);
}